// EmbeddedDecisionRules_79920751444158
// MI455X (gfx1250) — compile-verified
//
#include <hip/hip_runtime.h>
#include <hip/hip_bf16.h>
#include <cmath>

// ---------------------------------------------------------------------------
// Sizes (from the reference)
// ---------------------------------------------------------------------------
#define BATCH      16384
#define D_EMB      512
#define NUM_NODES  341          // 1 + 4 + 16 + 64 + 256
#define NC_REAL    1364         // NUM_NODES * 4 child prototypes
#define NC_PAD     1408         // padded to a multiple of 64 (wave N-tile)
#define N_GROUPS   22           // NC_PAD / 64
#define M_TILES32  512          // BATCH / 32  (32-row wave tiles)

// d_out layout (floats): logits | preds | probs | entropy
#define OFF_PREDS    22347776u  // 341*16384*4
#define OFF_PROBS    27934720u
#define OFF_ENTROPY  50282496u

// d_ws layout (floats): W[1408*512] | p2[1408] | x2[16384]
#define WS_P2  720896u
#define WS_X2  722304u

typedef __attribute__((ext_vector_type(2))) float v2f;
typedef __attribute__((ext_vector_type(8))) float v8f;

// ---------------------------------------------------------------------------
// Kernel A: child prototypes (masked means over contiguous leaf ranges) + ||p||^2
// One 256-thread block per padded column j; threads cover the 512-dim embedding.
// ---------------------------------------------------------------------------
__global__ __launch_bounds__(256) void build_w_kernel(
    const float* __restrict__ protos, float* __restrict__ W, float* __restrict__ p2) {
  const int j   = blockIdx.x;      // 0..NC_PAD-1
  const int tid = threadIdx.x;     // 0..255
  __shared__ float red[256];

  float w0 = 0.0f, w1 = 0.0f;
  if (j < NC_REAL) {
    const int node = j >> 2, c = j & 3;
    int L, start;
    if      (node >= 85) { L = 4; start = 85; }
    else if (node >= 21) { L = 3; start = 21; }
    else if (node >=  5) { L = 2; start =  5; }
    else if (node >=  1) { L = 1; start =  1; }
    else                 { L = 0; start =  0; }
    const int p    = node - start;
    const int span = 1024 >> (2 * L);
    const int cs   = span >> 2;                 // leaves under this child
    const int s    = p * span + c * cs;         // first leaf class
    const float inv = 1.0f / (float)cs;
    float a0 = 0.0f, a1 = 0.0f;
    for (int t = 0; t < cs; ++t) {
      const float* row = protos + (size_t)(s + t) * D_EMB;
      a0 += row[tid];
      a1 += row[tid + 256];
    }
    w0 = a0 * inv;
    w1 = a1 * inv;
  }
  W[(size_t)j * D_EMB + tid]       = w0;
  W[(size_t)j * D_EMB + tid + 256] = w1;

  red[tid] = w0 * w0 + w1 * w1;
  __syncthreads();
  for (int off = 128; off > 0; off >>= 1) {
    if (tid < off) red[tid] += red[tid + off];
    __syncthreads();
  }
  if (tid == 0) p2[j] = red[0];
}

// ---------------------------------------------------------------------------
// Kernel B: x2[b] = ||outputs[b]||^2 — one wave32 per row, float4 loads.
// ---------------------------------------------------------------------------
__global__ __launch_bounds__(256) void x2_kernel(
    const float* __restrict__ X, float* __restrict__ x2) {
  const int row  = (blockIdx.x * blockDim.x + threadIdx.x) >> 5;
  const int lane = threadIdx.x & 31;
  const float4* p = (const float4*)(X + (size_t)row * D_EMB + lane * 16);
  float s = 0.0f;
#pragma unroll
  for (int i = 0; i < 4; ++i) {
    float4 v = p[i];
    s += v.x * v.x + v.y * v.y + v.z * v.z + v.w * v.w;
  }
#pragma unroll
  for (int m = 16; m >= 1; m >>= 1) s += __shfl_xor(s, m);
  if (lane == 0) x2[row] = s;
}

// ---------------------------------------------------------------------------
// Kernel C: wave-tiled f32 WMMA GEMM, 32M x 64N per wave (8 accumulators),
// K stepped by 4. Each B fragment feeds two WMMAs (2 M sub-tiles); each A
// fragment feeds four (4 N tiles): 6 b64 loads -> 8 WMMAs per K-step.
// Fused epilogue: L2 distance -> log_softmax(4) -> probs/entropy/argmax.
// ---------------------------------------------------------------------------
__global__ __launch_bounds__(128) void hier_gemm_kernel(
    const float* __restrict__ X, const float* __restrict__ W,
    const float* __restrict__ p2, const float* __restrict__ x2,
    float* __restrict__ out) {
  const int lane   = threadIdx.x & 31;
  const int wave   = (blockIdx.x * 128 + threadIdx.x) >> 5;   // 0..11263 exact
  const int mTile  = wave / N_GROUPS;                         // 32-row tiles
  const int nGroup = wave - mTile * N_GROUPS;
  const int m0 = mTile << 5;
  const int n0 = nGroup << 6;
  const int half = lane >> 4;    // 0 or 1
  const int l15  = lane & 15;

  // A (16x4 f32) fragment: lane l -> row base+(l&15), K pair at k0 + 2*(l>>4)
  // B (4x16 f32) fragment: lane l -> col n0+(l&15), same K pair (row-striped)
  const float* aptr0 = X + (size_t)(m0 + l15) * D_EMB + (half << 1);
  const float* aptr1 = aptr0 + 16 * D_EMB;
  const float* bptr  = W + (size_t)(n0 + l15) * D_EMB + (half << 1);

  v8f acc[2][4];
#pragma unroll
  for (int mi = 0; mi < 2; ++mi)
#pragma unroll
    for (int t = 0; t < 4; ++t) acc[mi][t] = (v8f)(0.0f);

  for (int k0 = 0; k0 < D_EMB; k0 += 4) {
    v2f a0 = *(const v2f*)(aptr0 + k0);
    v2f a1 = *(const v2f*)(aptr1 + k0);
#pragma unroll
    for (int t = 0; t < 4; ++t) {
      v2f b = *(const v2f*)(bptr + k0 + t * 16 * D_EMB);
      acc[0][t] = __builtin_amdgcn_wmma_f32_16x16x4_f32(
          false, a0, false, b, (short)0, acc[0][t], false, false);
      acc[1][t] = __builtin_amdgcn_wmma_f32_16x16x4_f32(
          false, a1, false, b, (short)0, acc[1][t], false, false);
    }
  }

  float* logits  = out;
  float* preds   = out + OFF_PREDS;
  float* probs   = out + OFF_PROBS;
  float* entropy = out + OFF_ENTROPY;

#pragma unroll
  for (int mi = 0; mi < 2; ++mi) {
    const int mbase = m0 + mi * 16 + (half << 3);
    // ||x||^2 for the 8 batch rows this lane's accumulator slots cover
    float x2v[8];
#pragma unroll
    for (int v = 0; v < 8; ++v) x2v[v] = x2[mbase + v];

#pragma unroll
    for (int t = 0; t < 4; ++t) {
      const int j    = n0 + t * 16 + l15;   // global child-proto column
      const int node = j >> 2;
      const int c    = j & 3;
      const float p2t = p2[j];
#pragma unroll
      for (int v = 0; v < 8; ++v) {
        const int brow = mbase + v;         // batch row (C/D layout: M = v + 8*half)
        const float xp = acc[mi][t][v];
        const float d2 = fmaxf(x2v[v] + p2t - 2.0f * xp, 0.0f);
        const float logit = -sqrtf(d2);

        // softmax over the 4 children == 4 adjacent lanes (same batch row)
        float mx = fmaxf(logit, __shfl_xor(logit, 1));
        mx = fmaxf(mx, __shfl_xor(mx, 2));
        const float e = __expf(logit - mx);
        float s = e + __shfl_xor(e, 1);
        s += __shfl_xor(s, 2);
        const float lp   = logit - mx - __logf(s);
        const float prob = e / s;
        const float et   = -prob * lp;
        float ent = et + __shfl_xor(et, 1);
        ent += __shfl_xor(ent, 2);
        int cand = (logit == mx) ? c : 4;   // first-max tie break (argmax)
        int pr = min(cand, __shfl_xor(cand, 1));
        pr = min(pr, __shfl_xor(pr, 2));

        if (node < NUM_NODES) {
          const size_t oo = (size_t)node * (BATCH * 4) + (size_t)brow * 4 + c;
          logits[oo] = logit;
          probs[oo]  = prob;
          if (c == 0) {
            const size_t po = (size_t)node * BATCH + brow;
            preds[po]   = (float)pr;
            entropy[po] = ent;
          }
        }
      }
    }
  }
}

// ---------------------------------------------------------------------------
extern "C" void kernel_launch(void* const* d_in, const int* in_sizes, int n_in,
                              void* d_out, int out_size, void* d_ws, size_t ws_size,
                              hipStream_t stream) {
  const float* X = (const float*)d_in[0];   // outputs    [16384, 512]
  const float* P = (const float*)d_in[1];   // prototypes [1024, 512]
  // d_in[2]/d_in[3] (leaf_idx / leaf_mask) encode the complete 4-ary tree;
  // ranges are recomputed analytically in build_w_kernel.

  float* ws = (float*)d_ws;
  float* W  = ws;             // [1408, 512]
  float* p2 = ws + WS_P2;     // [1408]
  float* x2 = ws + WS_X2;     // [16384]
  float* out = (float*)d_out;

  build_w_kernel<<<NC_PAD, 256, 0, stream>>>(P, W, p2);
  x2_kernel<<<BATCH / 8, 256, 0, stream>>>(X, x2);
  hier_gemm_kernel<<<(M_TILES32 * N_GROUPS) / 4, 128, 0, stream>>>(X, W, p2, x2, out);
}